// Transformer_30511447671566
// MI455X (gfx1250) — compile-verified
//
#include <hip/hip_runtime.h>
#include <math.h>

// ---------------------------------------------------------------------------
// Transformer block for MI455X (gfx1250, wave32).
//   * GEMMs: V_WMMA_F32_16X16X4_F32 (fp32, matches reference numerics),
//     128x16 block tiles, LDS-staged operands, tile fills via the Tensor
//     Data Mover (tensor_load_to_lds + s_wait_tensorcnt) with TDM padding
//     to avoid LDS bank conflicts on A-column reads.
//   * Attention (HD=2, NH=T=384): flash-style online softmax in VALU with
//     K/V staged in LDS (matrix units useless at K=2).
// ---------------------------------------------------------------------------

typedef __attribute__((ext_vector_type(2))) float        v2f;
typedef __attribute__((ext_vector_type(8))) float        v8f;
typedef __attribute__((ext_vector_type(4))) unsigned int v4u;
typedef __attribute__((ext_vector_type(8))) int          v8i;
typedef __attribute__((ext_vector_type(4))) int          v4i;

#define B_   2
#define T_   384
#define C_   768
#define M_   (B_ * T_)          // 768 token rows
#define NH_  384
#define SCALE_ 0.70710678118654752f   // (1/sqrt(HD)) * sqrt(RANK), HD=2, RANK=1
#define LN_EPS_ 1e-5f

#define BM_   128               // rows per block (8 waves x 16)
#define KC_   64                // K chunk staged in LDS
#define APAD_ 2                 // dword pad per A row (bank-conflict break)

#if __has_builtin(__builtin_amdgcn_tensor_load_to_lds)
#define USE_TDM 1
#else
#define USE_TDM 0
#endif

__device__ __forceinline__ float gelu_exact(float v) {
    // approximate=False GELU: 0.5*x*(1+erf(x/sqrt(2)))
    return 0.5f * v * (1.0f + erff(v * 0.70710678118654752f));
}

#if USE_TDM
// ---------------------------------------------------------------------------
// Issue one TDM 2-D tile load (global -> LDS), ISA 8.3/8.4 descriptor layout.
//   group0: count=1 | lds_addr | global_addr[56:0] | type=2
//   group1: data_size=4B (+optional LDS pad fields) | tensor dims | tile dims
//           | tensor_dim0_stride (elements)
//   groups 2/3: zero (2-D tensor)
// pad_bits goes into group1 word0 (pad_enable<<20 | interval<<22 | amount<<25).
// This toolchain exposes the 6-arg builtin:
//   (v4u g0, v8i g1, v4i g2, v4i g3, v8i extra, i32 cpol)
// ---------------------------------------------------------------------------
__device__ __forceinline__ void tdm_load_2d(unsigned lds_byte_addr,
                                            const float* gptr,
                                            unsigned tile_d0, unsigned tile_d1,
                                            unsigned tensor_d0, unsigned tensor_d1,
                                            unsigned long long stride0_elems,
                                            unsigned pad_bits) {
    const unsigned long long ga = (unsigned long long)(uintptr_t)gptr;
    v4u g0;
    g0[0] = 1u;                                          // count=1, user D#
    g0[1] = lds_byte_addr;                               // lds_addr
    g0[2] = (unsigned)(ga & 0xffffffffu);                // global_addr[31:0]
    g0[3] = (unsigned)((ga >> 32) & 0x01ffffffu)         // global_addr[56:32]
          | (2u << 30);                                  // type=2 ("image")
    v8i g1;
    g1[0] = (int)((2u << 16) | pad_bits);                // data_size=4B (+pad)
    g1[1] = (int)((tensor_d0 & 0xffffu) << 16);          // tensor_dim0[15:0]
    g1[2] = (int)(((tensor_d0 >> 16) & 0xffffu)
          |        ((tensor_d1 & 0xffffu) << 16));       // dim0[31:16]|dim1[15:0]
    g1[3] = (int)(((tensor_d1 >> 16) & 0xffffu)
          |        ((tile_d0 & 0xffffu) << 16));         // dim1[31:16]|tile_dim0
    g1[4] = (int)(tile_d1 & 0xffffu);                    // tile_dim1 (tile_dim2=0)
    g1[5] = (int)(unsigned)(stride0_elems & 0xffffffffu);// dim0_stride[31:0]
    g1[6] = (int)(unsigned)((stride0_elems >> 32) & 0xffffu); // stride[47:32]
    g1[7] = 0;                                           // dim1_stride unused (2D)
    const v4i z4 = {0, 0, 0, 0};
    const v8i z8 = {0, 0, 0, 0, 0, 0, 0, 0};
    __builtin_amdgcn_tensor_load_to_lds(g0, g1, z4, z4, z8, 0);
}
#endif

// ---------------------------------------------------------------------------
// LayerNorm: one 256-thread block per row of 768 elements.
// ---------------------------------------------------------------------------
__global__ __launch_bounds__(256) void ln_kernel(const float* __restrict__ x,
                                                 const float* __restrict__ g,
                                                 const float* __restrict__ b,
                                                 float* __restrict__ out) {
    __shared__ float red[256];
    const int row = blockIdx.x;
    const int tid = threadIdx.x;
    const float* xr = x + (size_t)row * C_;

    float loc[3];
    float s = 0.f;
#pragma unroll
    for (int j = 0; j < 3; ++j) { loc[j] = xr[tid + 256 * j]; s += loc[j]; }
    red[tid] = s;
    __syncthreads();
    for (int off = 128; off > 0; off >>= 1) {
        if (tid < off) red[tid] += red[tid + off];
        __syncthreads();
    }
    const float mean = red[0] * (1.0f / (float)C_);
    __syncthreads();

    float v = 0.f;
#pragma unroll
    for (int j = 0; j < 3; ++j) { float d = loc[j] - mean; v += d * d; }
    red[tid] = v;
    __syncthreads();
    for (int off = 128; off > 0; off >>= 1) {
        if (tid < off) red[tid] += red[tid + off];
        __syncthreads();
    }
    const float rstd = rsqrtf(red[0] * (1.0f / (float)C_) + LN_EPS_);

#pragma unroll
    for (int j = 0; j < 3; ++j) {
        const int i = tid + 256 * j;
        out[(size_t)row * C_ + i] = (loc[j] - mean) * rstd * g[i] + b[i];
    }
}

// ---------------------------------------------------------------------------
// fp32 WMMA GEMM: Y[M,N] = A[M,K] @ W[K,N] + bias[N]  (+ epilogue)
//   epilogue 0: none;  1: += R;  2: = R + gelu(acc+bias)
// Block = 256 threads (8 waves), tile = 128(M) x 16(N); K chunked by 64.
// Wave 0 fills the A/B LDS tiles with two TDM descriptor loads (A with TDM
// LDS-padding: 2 dwords after every 64-dword row), waits TENSORcnt, then the
// workgroup barrier releases 8 waves of pure ds_load + v_wmma_f32_16x16x4_f32.
//
// WMMA operand layout (ISA 7.12.2): A 16x4 f32 -> lanes 0-15 hold {K,K+1} of
// row M=lane, lanes 16-31 hold {K+2,K+3}; B mirrored; C/D: VGPR r holds rows
// r (lanes 0-15) and r+8 (lanes 16-31), N = lane&15.
// ---------------------------------------------------------------------------
__global__ __launch_bounds__(256) void gemm_wmma_f32(const float* __restrict__ A,
                                                     const float* __restrict__ W,
                                                     const float* __restrict__ bias,
                                                     const float* __restrict__ R,
                                                     float* __restrict__ Y,
                                                     int M, int K, int N, int epilogue) {
    __shared__ float At[BM_ * (KC_ + APAD_)];   // 128 x (64+2) f32 = 33 KB
    __shared__ float Bt[KC_ * 16];              // 64 x 16 f32      =  4 KB

    const int tid  = threadIdx.x;
    const int wave = tid >> 5;
    const int lane = tid & 31;
    const int half = lane >> 4;
    const int l16  = lane & 15;
    const int n0    = blockIdx.x * 16;
    const int mBase = blockIdx.y * BM_;

    v8f acc = {};

    for (int k0 = 0; k0 < K; k0 += KC_) {
#if USE_TDM
        if (wave == 0) {
            // A tile: 128 rows x 64 cols, row stride K, padded in LDS:
            // pad_enable(1)<<20 | pad_interval(64 dw -> code 5)<<22
            // | pad_amount(2 dw -> code 1)<<25
            tdm_load_2d((unsigned)(uintptr_t)(void*)At,
                        A + (size_t)mBase * K + k0,
                        /*tile*/ KC_, BM_, /*tensor*/ (unsigned)K, (unsigned)M,
                        (unsigned long long)K,
                        (1u << 20) | (5u << 22) | (1u << 25));
            // B tile: 64 rows x 16 cols, row stride N, no padding
            tdm_load_2d((unsigned)(uintptr_t)(void*)Bt,
                        W + (size_t)k0 * N + n0,
                        /*tile*/ 16, KC_, /*tensor*/ (unsigned)N, (unsigned)K,
                        (unsigned long long)N, 0u);
            __builtin_amdgcn_s_wait_tensorcnt(0);
        }
#else
        for (int idx = tid; idx < BM_ * KC_; idx += 256) {
            const int r = idx / KC_, c = idx % KC_;
            At[r * (KC_ + APAD_) + c] = A[(size_t)(mBase + r) * K + k0 + c];
        }
        for (int idx = tid; idx < KC_ * 16; idx += 256) {
            const int r = idx >> 4, c = idx & 15;
            Bt[r * 16 + c] = W[(size_t)(k0 + r) * N + n0 + c];
        }
#endif
        // Prefetch next B chunk toward L2 while this one is consumed.
        if (wave == 1 && k0 + KC_ < K)
            __builtin_prefetch(&W[(size_t)(k0 + KC_ + half) * N + n0 + l16], 0, 3);
        __syncthreads();

        const float* at = &At[(wave * 16 + l16) * (KC_ + APAD_) + half * 2];
        const float* bt = &Bt[(half * 2) * 16 + l16];
#pragma unroll
        for (int kk = 0; kk < KC_; kk += 4) {
            v2f a, b;
            a.x = at[0];            // ds_load_b64 (8B aligned: even-dword col)
            a.y = at[1];
            b.x = bt[0];
            b.y = bt[16];
            acc = __builtin_amdgcn_wmma_f32_16x16x4_f32(
                /*neg_a=*/false, a, /*neg_b=*/false, b,
                /*c_mod=*/(short)0, acc, /*reuse_a=*/false, /*reuse_b=*/false);
            at += 4;
            bt += 4 * 16;
        }
        __syncthreads();
    }

    const int col = n0 + l16;
    const float bv = bias[col];
#pragma unroll
    for (int r = 0; r < 8; ++r) {
        const int row = mBase + wave * 16 + r + half * 8;
        float val = acc[r] + bv;
        if (epilogue == 1) {
            val += R[(size_t)row * N + col];
        } else if (epilogue == 2) {
            val = R[(size_t)row * N + col] + gelu_exact(val);
        }
        Y[(size_t)row * N + col] = val;
    }
}

// ---------------------------------------------------------------------------
// Attention over the HEAD axis, per (b,t): scores NH x NH = 384x384 from
// q,k in [384,2]; causal mask; softmax over m; y = att @ v.  One block per
// (b,t); 384 threads (one per head n); K/V in LDS; online softmax.
// qkv row layout: [q(768) | k(768) | v(768)], head n at elements 2n, 2n+1.
// ---------------------------------------------------------------------------
__global__ __launch_bounds__(NH_) void attn_kernel(const float* __restrict__ qkv,
                                                   float* __restrict__ y) {
    __shared__ float2 ks[NH_];
    __shared__ float2 vs[NH_];

    const int bt = blockIdx.x;                         // 0 .. B*T-1
    const int n  = threadIdx.x;                        // head index 0..383
    const float* base = qkv + (size_t)bt * (3 * C_);

    ks[n] = make_float2(base[C_     + 2 * n], base[C_     + 2 * n + 1]);
    vs[n] = make_float2(base[2 * C_ + 2 * n], base[2 * C_ + 2 * n + 1]);
    __syncthreads();

    const float qx = base[2 * n];
    const float qy = base[2 * n + 1];

    float mx = -INFINITY;   // running max
    float l  = 0.f;         // running denom
    float ax = 0.f, ay = 0.f;
    for (int m = 0; m <= n; ++m) {             // causal: m <= n
        const float2 kv = ks[m];
        const float s  = (qx * kv.x + qy * kv.y) * SCALE_;
        const float nm = fmaxf(mx, s);
        const float corr = __expf(mx - nm);    // first iter: exp(-inf)=0
        const float p    = __expf(s - nm);
        l  = l * corr + p;
        const float2 vv = vs[m];
        ax = ax * corr + p * vv.x;
        ay = ay * corr + p * vv.y;
        mx = nm;
    }
    const float inv = 1.0f / l;
    y[(size_t)bt * C_ + 2 * n]     = ax * inv;
    y[(size_t)bt * C_ + 2 * n + 1] = ay * inv;
}

// ---------------------------------------------------------------------------
// Host-side orchestration (graph-capture safe: launches only, scratch in d_ws)
// ---------------------------------------------------------------------------
extern "C" void kernel_launch(void* const* d_in, const int* in_sizes, int n_in,
                              void* d_out, int out_size, void* d_ws, size_t ws_size,
                              hipStream_t stream) {
    (void)in_sizes; (void)n_in; (void)out_size; (void)ws_size;

    const float* x      = (const float*)d_in[0];
    const float* W_qkv  = (const float*)d_in[1];
    const float* b_qkv  = (const float*)d_in[2];
    const float* W_proj = (const float*)d_in[3];
    const float* b_proj = (const float*)d_in[4];
    const float* ln1_g  = (const float*)d_in[5];
    const float* ln1_b  = (const float*)d_in[6];
    const float* ln2_g  = (const float*)d_in[7];
    const float* ln2_b  = (const float*)d_in[8];
    const float* W1     = (const float*)d_in[9];
    const float* b1     = (const float*)d_in[10];
    const float* W2     = (const float*)d_in[11];
    const float* b2     = (const float*)d_in[12];
    float* out = (float*)d_out;

    // Workspace layout (floats). Total = 768*8448 floats ~ 24.8 MiB.
    float* ws = (float*)d_ws;
    size_t off = 0;
    float* h    = ws + off; off += (size_t)M_ * C_;        // LN1 output
    float* qkv  = ws + off; off += (size_t)M_ * 3 * C_;    // qkv
    float* yb   = ws + off; off += (size_t)M_ * C_;        // attention output
    float* x1   = ws + off; off += (size_t)M_ * C_;        // residual after attn
    float* h2   = ws + off; off += (size_t)M_ * C_;        // LN2 output
    float* u    = ws + off; off += (size_t)M_ * 4 * C_;    // FF hidden

    // 1) h = LN1(x)
    ln_kernel<<<M_, 256, 0, stream>>>(x, ln1_g, ln1_b, h);

    // 2) qkv = h @ W_qkv + b_qkv          [768 x 2304]
    gemm_wmma_f32<<<dim3((3 * C_) / 16, M_ / BM_), 256, 0, stream>>>(
        h, W_qkv, b_qkv, nullptr, qkv, M_, C_, 3 * C_, 0);

    // 3) attention per (b,t)
    attn_kernel<<<M_, NH_, 0, stream>>>(qkv, yb);

    // 4) x1 = x + yb @ W_proj + b_proj
    gemm_wmma_f32<<<dim3(C_ / 16, M_ / BM_), 256, 0, stream>>>(
        yb, W_proj, b_proj, x, x1, M_, C_, C_, 1);

    // 5) h2 = LN2(x1)
    ln_kernel<<<M_, 256, 0, stream>>>(x1, ln2_g, ln2_b, h2);

    // 6) u = h2 @ W1 + b1                 [768 x 3072]
    gemm_wmma_f32<<<dim3((4 * C_) / 16, M_ / BM_), 256, 0, stream>>>(
        h2, W1, b1, nullptr, u, M_, C_, 4 * C_, 0);

    // 7) out = x1 + gelu(u @ W2 + b2)
    gemm_wmma_f32<<<dim3(C_ / 16, M_ / BM_), 256, 0, stream>>>(
        u, W2, b2, x1, out, M_, 4 * C_, C_, 2);
}